// PPO_52965536694970
// MI455X (gfx1250) — compile-verified
//
#include <hip/hip_runtime.h>
#include <hip/hip_bf16.h>
#include <math.h>

#define T_HOR 65536
#define S_DIM 256
#define A_DIM 64
#define HID 256
#define GAMMA 0.99f
#define LMBDA 0.95f
#define EPS_CLIP 0.2f
#define VF_COEF 0.5f
#define ENT_COEF 0.01f
#define LOG2PI 1.8378770664093453f

typedef __attribute__((ext_vector_type(2))) float v2f;
typedef __attribute__((ext_vector_type(8))) float v8f;
typedef __attribute__((ext_vector_type(4))) float f4v;
typedef __attribute__((ext_vector_type(4))) int   v4i;

typedef __attribute__((address_space(1))) v4i*  g_v4i_p;
typedef __attribute__((address_space(3))) v4i*  l_v4i_p;
typedef const __attribute__((address_space(1))) char* g_chr_p;
typedef __attribute__((address_space(3))) char* l_chr_p;

// Async global->LDS path (gfx1250): guarded so either toolchain compiles.
#if defined(__has_builtin)
#if __has_builtin(__builtin_amdgcn_global_load_async_to_lds_b128)
#define HAVE_ASYNC_LDS 1
#endif
#endif

__device__ __forceinline__ void wait_async_zero() {
#if defined(__has_builtin)
#if __has_builtin(__builtin_amdgcn_s_wait_asynccnt)
  __builtin_amdgcn_s_wait_asynccnt(0);
  return;
#else
  asm volatile("s_wait_asynccnt 0x0" ::: "memory");
  return;
#endif
#else
  asm volatile("s_wait_asynccnt 0x0" ::: "memory");
#endif
}

// LDS layout (floats) for fwd_kernel (dynamic shared, ~193 KB of 320 KB/WGP)
#define OFF_S    0          // 64x256 staged s tile
#define OFF_HA   16384      // 64x256 hidden buffer A
#define OFF_HB   32768      // 64x256 hidden buffer B
#define OFF_LSTD 49152      // 64 log_std
#define OFF_ESTD 49216      // 64 exp(-log_std)
#define OFF_MUB  49280      // 64 mu bias
#define OFF_RED  49344      // 128 reduction scratch
#define SMEM_FLOATS 49472

#define WMMA4(acc, af, bf)                                                   \
  acc = __builtin_amdgcn_wmma_f32_16x16x4_f32(false, af, false, bf, (short)0, \
                                              acc, false, false)

// ---------------------------------------------------------------------------
// 64x256 @ 256x256 GEMM (+bias, relu) with V_WMMA_F32_16X16X4_F32.
// 4 waves; each wave owns a 16-row tile and 4 column tiles at a time:
// one A fragment feeds 4 independent WMMA accumulator chains (A reuse, and
// independent chains so WMMAs pipeline without RAW stalls).
// ---------------------------------------------------------------------------
__device__ __forceinline__ void gemm_relu_256(const float* __restrict__ Alds,
                                              const float* __restrict__ W,
                                              const float* __restrict__ bias,
                                              float* __restrict__ Out) {
  const int tid  = threadIdx.x;
  const int wave = tid >> 5;
  const int lane = tid & 31;
  const int l16  = lane & 15;
  const int hi   = lane >> 4;   // 0 or 1
  const int kh   = hi << 1;     // K half offset (0 or 2)
  const int rOff = hi << 3;     // C/D row offset (0 or 8)
  const int m0   = wave << 4;   // 16-row tile per wave
  const float* arowBase = Alds + (m0 + l16) * HID + kh;

  for (int n0 = 0; n0 < HID; n0 += 64) {
    v8f a0 = {0.f, 0.f, 0.f, 0.f, 0.f, 0.f, 0.f, 0.f};
    v8f a1 = a0, a2 = a0, a3 = a0;
    const float* wcol = W + kh * HID + n0 + l16;
#pragma unroll 2
    for (int k0 = 0; k0 < HID; k0 += 4) {
      v2f af;
      af.x = arowBase[k0];
      af.y = arowBase[k0 + 1];
      const float* wk = wcol + k0 * HID;
      v2f b0, b1, b2, b3;
      b0.x = wk[0];       b0.y = wk[HID];
      b1.x = wk[16];      b1.y = wk[HID + 16];
      b2.x = wk[32];      b2.y = wk[HID + 32];
      b3.x = wk[48];      b3.y = wk[HID + 48];
      WMMA4(a0, af, b0);
      WMMA4(a1, af, b1);
      WMMA4(a2, af, b2);
      WMMA4(a3, af, b3);
    }
    float bs0 = bias[n0 + l16];
    float bs1 = bias[n0 + 16 + l16];
    float bs2 = bias[n0 + 32 + l16];
    float bs3 = bias[n0 + 48 + l16];
    float* orow = Out + (m0 + rOff) * HID + n0 + l16;
#pragma unroll
    for (int r = 0; r < 8; ++r) {
      orow[r * HID +  0] = fmaxf(a0[r] + bs0, 0.f);
      orow[r * HID + 16] = fmaxf(a1[r] + bs1, 0.f);
      orow[r * HID + 32] = fmaxf(a2[r] + bs2, 0.f);
      orow[r * HID + 48] = fmaxf(a3[r] + bs3, 0.f);
    }
  }
}

// ---------------------------------------------------------------------------
// Fused forward: value net -> vals[t]; policy net -> log_prob[t].
// Reads s exactly once (async-staged to LDS). grid = T/64, block = 128.
// ---------------------------------------------------------------------------
__global__ void fwd_kernel(const float* __restrict__ s, const float* __restrict__ act,
                           const float* __restrict__ piW1, const float* __restrict__ piB1,
                           const float* __restrict__ piW2, const float* __restrict__ piB2,
                           const float* __restrict__ muW,  const float* __restrict__ muB,
                           const float* __restrict__ logstd,
                           const float* __restrict__ vW1, const float* __restrict__ vB1,
                           const float* __restrict__ vW2, const float* __restrict__ vB2,
                           const float* __restrict__ vhW, const float* __restrict__ vhB,
                           float* __restrict__ vals, float* __restrict__ lp) {
  extern __shared__ float sm[];
  float* sT   = sm + OFF_S;
  float* hA   = sm + OFF_HA;
  float* hB   = sm + OFF_HB;
  float* lstd = sm + OFF_LSTD;
  float* estd = sm + OFF_ESTD;
  float* mub  = sm + OFF_MUB;
  float* red  = sm + OFF_RED;

  const int tid = threadIdx.x;
  const size_t rowBase = (size_t)blockIdx.x * 64;

  // Stage s tile 64x256 into LDS.
#if HAVE_ASYNC_LDS
  {
    g_chr_p g = (g_chr_p)(s + rowBase * S_DIM);
    l_chr_p l = (l_chr_p)sT;
    // 64*256*4 = 64 KB; 128 lanes x 16 B = 2 KB per wave-instruction pass.
    for (int off = tid * 16; off < 64 * S_DIM * 4; off += 128 * 16) {
      __builtin_amdgcn_global_load_async_to_lds_b128((g_v4i_p)(g + off),
                                                     (l_v4i_p)(l + off), 0, 0);
    }
    wait_async_zero();
  }
#else
  {
    const f4v* s4 = (const f4v*)(s + rowBase * S_DIM);
    f4v* d4 = (f4v*)sT;
    for (int i = tid; i < 64 * S_DIM / 4; i += 128) d4[i] = s4[i];
  }
#endif
  // Prefetch next block's s tile into cache while we compute.
  if (blockIdx.x + 1 < gridDim.x) {
#pragma unroll
    for (int i = 0; i < 4; ++i)
      __builtin_prefetch(s + (rowBase + 64) * S_DIM + (size_t)(tid + 128 * i) * 32, 0, 1);
  }
  if (tid < A_DIM) {
    float ls = logstd[tid];
    lstd[tid] = ls;
    estd[tid] = __expf(-ls);
    mub[tid]  = muB[tid];
  }
  __syncthreads();

  // ---------------- value net ----------------
  gemm_relu_256(sT, vW1, vB1, hA);
  __syncthreads();
  gemm_relu_256(hA, vW2, vB2, hB);
  __syncthreads();
  {  // value head: 2 threads per row, skewed LDS reads to spread banks
    int row = tid & 63;
    int k0  = (tid >> 6) << 7;  // 0 or 128
    float p = 0.f;
    const float* hrow = hB + row * HID;
    for (int i = 0; i < 128; ++i) {
      int k = k0 + ((i + row) & 127);
      p += hrow[k] * vhW[k];
    }
    red[tid] = p;
    __syncthreads();
    if (tid < 64) vals[rowBase + tid] = red[tid] + red[tid + 64] + vhB[0];
    __syncthreads();
  }

  // ---------------- policy net ----------------
  gemm_relu_256(sT, piW1, piB1, hA);
  __syncthreads();
  gemm_relu_256(hA, piW2, piB2, hB);
  __syncthreads();

  // mu head (64x256 @ 256x64) fused with Gaussian log-prob; mu never hits HBM
  {
    const int wave = tid >> 5;
    const int lane = tid & 31;
    const int l16  = lane & 15;
    const int hi   = lane >> 4;
    const int kh   = hi << 1;
    const int rOff = hi << 3;
    const int m0   = wave << 4;

    v8f a0 = {0.f, 0.f, 0.f, 0.f, 0.f, 0.f, 0.f, 0.f};
    v8f a1 = a0, a2 = a0, a3 = a0;
    const float* arowBase = hB + (m0 + l16) * HID + kh;
    const float* wcol = muW + kh * A_DIM + l16;
#pragma unroll 2
    for (int k0 = 0; k0 < HID; k0 += 4) {
      v2f af;
      af.x = arowBase[k0];
      af.y = arowBase[k0 + 1];
      const float* wk = wcol + k0 * A_DIM;
      v2f b0, b1, b2, b3;
      b0.x = wk[0];       b0.y = wk[A_DIM];
      b1.x = wk[16];      b1.y = wk[A_DIM + 16];
      b2.x = wk[32];      b2.y = wk[A_DIM + 32];
      b3.x = wk[48];      b3.y = wk[A_DIM + 48];
      WMMA4(a0, af, b0);
      WMMA4(a1, af, b1);
      WMMA4(a2, af, b2);
      WMMA4(a3, af, b3);
    }
    float lpa[8] = {0.f, 0.f, 0.f, 0.f, 0.f, 0.f, 0.f, 0.f};
    int n0 = l16, n1 = 16 + l16, n2 = 32 + l16, n3 = 48 + l16;
#pragma unroll
    for (int r = 0; r < 8; ++r) {
      size_t trow = rowBase + (size_t)(m0 + rOff + r);
      const float* arow = act + trow * A_DIM;
      float z0 = (arow[n0] - (a0[r] + mub[n0])) * estd[n0];
      float z1 = (arow[n1] - (a1[r] + mub[n1])) * estd[n1];
      float z2 = (arow[n2] - (a2[r] + mub[n2])) * estd[n2];
      float z3 = (arow[n3] - (a3[r] + mub[n3])) * estd[n3];
      lpa[r] = -0.5f * (z0 * z0 + z1 * z1 + z2 * z2 + z3 * z3)
               - (lstd[n0] + lstd[n1] + lstd[n2] + lstd[n3]);
    }
    // horizontal sum over the 16 lanes holding each row's A_DIM slices
#pragma unroll
    for (int r = 0; r < 8; ++r) {
      float v = lpa[r];
      v += __shfl_xor(v, 1, 16);
      v += __shfl_xor(v, 2, 16);
      v += __shfl_xor(v, 4, 16);
      v += __shfl_xor(v, 8, 16);
      if (l16 == 0)
        lp[rowBase + (size_t)(m0 + rOff + r)] = v - 0.5f * LOG2PI * (float)A_DIM;
    }
  }
}

// ---------------------------------------------------------------------------
// Single-row value net for sp_last (VALU; negligible cost)
// ---------------------------------------------------------------------------
__global__ void vlast_kernel(const float* __restrict__ sp,
                             const float* __restrict__ vW1, const float* __restrict__ vB1,
                             const float* __restrict__ vW2, const float* __restrict__ vB2,
                             const float* __restrict__ vhW, const float* __restrict__ vhB,
                             float* __restrict__ vlast) {
  __shared__ float s0[S_DIM];
  __shared__ float h1[HID];
  __shared__ float red[256];
  int tid = threadIdx.x;
  s0[tid] = sp[tid];
  __syncthreads();
  float acc = vB1[tid];
  for (int k = 0; k < S_DIM; ++k) acc += s0[k] * vW1[k * HID + tid];
  h1[tid] = fmaxf(acc, 0.f);
  __syncthreads();
  acc = vB2[tid];
  for (int k = 0; k < HID; ++k) acc += h1[k] * vW2[k * HID + tid];
  float h2 = fmaxf(acc, 0.f);
  red[tid] = h2 * vhW[tid];
  __syncthreads();
  for (int s = 128; s > 0; s >>= 1) {
    if (tid < s) red[tid] += red[tid + s];
    __syncthreads();
  }
  if (tid == 0) vlast[0] = red[0] + vhB[0];
}

// ---------------------------------------------------------------------------
// GAE: chunked linear-recurrence scan. adv[t] = d[t] + (g*l*nd[t]) * adv[t+1].
// 256 threads x 256-element chunks; (A,B) summaries; serial carry over 256
// chunks; re-apply. Also computes mean / std(ddof=1) of adv.
// ---------------------------------------------------------------------------
__global__ void gae_kernel(const float* __restrict__ r, const float* __restrict__ nd,
                           const float* __restrict__ vals, const float* __restrict__ vlast,
                           float* __restrict__ adv, float* __restrict__ stats) {
  __shared__ float Aj[256], Bj[256], inc[256];
  __shared__ float rs1[256], rs2[256];
  const int tid = threadIdx.x;
  const int C = 256;
  const int base = tid * C;
  const float vl = vlast[0];
  const float gl = GAMMA * LMBDA;

  float Ac = 1.f, Bc = 0.f;
  for (int i = C - 1; i >= 0; --i) {
    int t = base + i;
    float ndt = nd[t];
    float vnext = (t == T_HOR - 1) ? vl : vals[t + 1];
    float d = r[t] + GAMMA * vnext * ndt - vals[t];
    float c = gl * ndt;
    Bc = d + c * Bc;
    Ac *= c;
  }
  Aj[tid] = Ac;
  Bj[tid] = Bc;
  __syncthreads();
  if (tid == 0) {
    float carry = 0.f;
    for (int j = 255; j >= 0; --j) {
      inc[j] = carry;
      carry = Bj[j] + Aj[j] * carry;
    }
  }
  __syncthreads();
  float carry = inc[tid];
  float s1 = 0.f, s2 = 0.f;
  for (int i = C - 1; i >= 0; --i) {
    int t = base + i;
    float ndt = nd[t];
    float vnext = (t == T_HOR - 1) ? vl : vals[t + 1];
    float d = r[t] + GAMMA * vnext * ndt - vals[t];
    carry = d + gl * ndt * carry;
    adv[t] = carry;
    s1 += carry;
    s2 += carry * carry;
  }
  rs1[tid] = s1;
  rs2[tid] = s2;
  __syncthreads();
  for (int s = 128; s > 0; s >>= 1) {
    if (tid < s) { rs1[tid] += rs1[tid + s]; rs2[tid] += rs2[tid + s]; }
    __syncthreads();
  }
  if (tid == 0) {
    float mean = rs1[0] / (float)T_HOR;
    float var = (rs2[0] - (float)T_HOR * mean * mean) / (float)(T_HOR - 1);
    stats[0] = mean;
    stats[1] = sqrtf(fmaxf(var, 0.f));
  }
}

// ---------------------------------------------------------------------------
// Per-element PPO loss terms; deterministic two-stage reduction.
// Note: v_pred == vals numerically, so v_pred - returns = -adv_raw.
// ---------------------------------------------------------------------------
__global__ void loss_kernel(const float* __restrict__ adv, const float* __restrict__ lp,
                            const float* __restrict__ lpold, const float* __restrict__ stats,
                            float* __restrict__ ppg, float* __restrict__ pv) {
  __shared__ float r1[256], r2[256];
  int tid = threadIdx.x;
  int t = blockIdx.x * 256 + tid;
  float mean = stats[0], sd = stats[1];
  float a = adv[t];
  float an = (a - mean) / (sd + 1e-8f);
  float ratio = __expf(lp[t] - lpold[t]);
  float rc = fminf(fmaxf(ratio, 1.f - EPS_CLIP), 1.f + EPS_CLIP);
  float pg = -fminf(ratio * an, rc * an);
  float d = -a;
  float ad = fabsf(d);
  float vl = (ad < 1.f) ? 0.5f * d * d : ad - 0.5f;
  r1[tid] = pg;
  r2[tid] = vl;
  __syncthreads();
  for (int s = 128; s > 0; s >>= 1) {
    if (tid < s) { r1[tid] += r1[tid + s]; r2[tid] += r2[tid + s]; }
    __syncthreads();
  }
  if (tid == 0) { ppg[blockIdx.x] = r1[0]; pv[blockIdx.x] = r2[0]; }
}

__global__ void combine_kernel(const float* __restrict__ ppg, const float* __restrict__ pv,
                               const float* __restrict__ logstd, float* __restrict__ out) {
  __shared__ float r1[256], r2[256], r3[256];
  int tid = threadIdx.x;
  r1[tid] = ppg[tid];
  r2[tid] = pv[tid];
  r3[tid] = (tid < A_DIM) ? logstd[tid] : 0.f;
  __syncthreads();
  for (int s = 128; s > 0; s >>= 1) {
    if (tid < s) { r1[tid] += r1[tid + s]; r2[tid] += r2[tid + s]; r3[tid] += r3[tid + s]; }
    __syncthreads();
  }
  if (tid == 0) {
    float pg = r1[0] / (float)T_HOR;
    float vloss = r2[0] / (float)T_HOR;
    float ent = 0.5f + 0.5f * LOG2PI + r3[0] / (float)A_DIM;
    out[0] = pg + VF_COEF * vloss - ENT_COEF * ent;
  }
}

// ---------------------------------------------------------------------------
extern "C" void kernel_launch(void* const* d_in, const int* in_sizes, int n_in,
                              void* d_out, int out_size, void* d_ws, size_t ws_size,
                              hipStream_t stream) {
  (void)in_sizes; (void)n_in; (void)out_size; (void)ws_size;
  const float* s      = (const float*)d_in[0];
  const float* a      = (const float*)d_in[1];
  const float* r      = (const float*)d_in[2];
  const float* sp     = (const float*)d_in[3];
  const float* lpold  = (const float*)d_in[4];
  const float* nd     = (const float*)d_in[5];
  const float* piW1   = (const float*)d_in[6];
  const float* piB1   = (const float*)d_in[7];
  const float* piW2   = (const float*)d_in[8];
  const float* piB2   = (const float*)d_in[9];
  const float* muW    = (const float*)d_in[10];
  const float* muB    = (const float*)d_in[11];
  const float* logstd = (const float*)d_in[12];
  const float* vW1    = (const float*)d_in[13];
  const float* vB1    = (const float*)d_in[14];
  const float* vW2    = (const float*)d_in[15];
  const float* vB2    = (const float*)d_in[16];
  const float* vhW    = (const float*)d_in[17];
  const float* vhB    = (const float*)d_in[18];

  float* ws    = (float*)d_ws;
  float* vals  = ws;
  float* lp    = ws + T_HOR;
  float* adv   = ws + 2 * T_HOR;
  float* vlast = ws + 3 * T_HOR;
  float* stats = ws + 3 * T_HOR + 4;
  float* ppg   = ws + 3 * T_HOR + 8;
  float* pv    = ws + 3 * T_HOR + 8 + 256;

  size_t shmem = (size_t)SMEM_FLOATS * sizeof(float);
  hipLaunchKernelGGL(fwd_kernel, dim3(T_HOR / 64), dim3(128), shmem, stream,
                     s, a, piW1, piB1, piW2, piB2, muW, muB, logstd,
                     vW1, vB1, vW2, vB2, vhW, vhB, vals, lp);
  hipLaunchKernelGGL(vlast_kernel, dim3(1), dim3(256), 0, stream,
                     sp, vW1, vB1, vW2, vB2, vhW, vhB, vlast);
  hipLaunchKernelGGL(gae_kernel, dim3(1), dim3(256), 0, stream,
                     r, nd, vals, vlast, adv, stats);
  hipLaunchKernelGGL(loss_kernel, dim3(256), dim3(256), 0, stream,
                     adv, lp, lpold, stats, ppg, pv);
  hipLaunchKernelGGL(combine_kernel, dim3(1), dim3(256), 0, stream,
                     ppg, pv, logstd, (float*)d_out);
}